// GRU_RGCN_30932354466392
// MI455X (gfx1250) — compile-verified
//
#include <hip/hip_runtime.h>

// Problem constants (from reference)
#define S_TOT 128   // B*T
#define NN    32    // grapharea size
#define DD    300   // feature dim
#define DP    304   // padded to 19*16 for WMMA N tiles
#define RR    4     // relations
#define EE    64    // edges per step
#define VV    50000 // vocab (= 3125 * 16, tile-exact)

typedef float v2f __attribute__((ext_vector_type(2)));
typedef float v4f __attribute__((ext_vector_type(4)));
typedef float v8f __attribute__((ext_vector_type(8)));
typedef int   i32x4 __attribute__((ext_vector_type(4)));

#define AS1 __attribute__((address_space(1)))
#define AS3 __attribute__((address_space(3)))

#if defined(__has_builtin)
#if __has_builtin(__builtin_amdgcn_global_load_async_to_lds_b128) && \
    __has_builtin(__builtin_amdgcn_s_wait_asynccnt)
#define USE_ASYNC_GATHER 1
#endif
#endif

// -----------------------------------------------------------------------------
// Pre-pass: pad the 5 [300x300] weight matrices (conv_W[0..3], W_0) into
// zero-padded [300x304] copies so the WMMA inner loop needs no column guards.
// -----------------------------------------------------------------------------
__global__ __launch_bounds__(256) void pad_weights_kernel(
    const float* __restrict__ conv_W, const float* __restrict__ W0,
    float* __restrict__ Wpad) {
  int p = blockIdx.x * blockDim.x + threadIdx.x;
  if (p >= 5 * DD * DP) return;
  int m = p / (DD * DP);
  int rem = p % (DD * DP);
  int k = rem / DP, d = rem % DP;
  const float* src = (m < RR) ? (conv_W + (long)m * DD * DD) : W0;
  Wpad[p] = (d < DD) ? src[k * DD + d] : 0.0f;
}

// -----------------------------------------------------------------------------
// FP32 WMMA helper: D(32xDP LDS) = A(32xDP LDS) * B(300xDP padded, global).
// Tiles: 2 M-tiles x 19 N-tiles over 32 waves, V_WMMA_F32_16X16X4_F32, K=300.
// A layout : lane<16 -> M=lane, K={k,k+1}; lane>=16 -> M=lane-16, K={k+2,k+3}
// B layout : lane<16 -> N=lane, K={k,k+1}; lane>=16 -> N=lane-16, K={k+2,k+3}
// C/D      : vgpr v, lane<16 -> (M=v, N=lane); lane>=16 -> (M=v+8, N=lane-16)
// -----------------------------------------------------------------------------
__device__ __forceinline__ void gemm_tiles_f32(const float (*A)[DP],
                                               const float* __restrict__ Bp,
                                               float (*Dst)[DP],
                                               int lane, int wid) {
  for (int tile = wid; tile < 2 * 19; tile += 32) {
    const int mt = tile / 19;
    const int nt = tile % 19;
    const int n0 = nt * 16;
    const int arow = mt * 16 + (lane & 15);
    const int bcol = n0 + (lane & 15);
    const int koff = (lane >> 4) << 1;
    v8f acc = {};
    for (int k = 0; k < DD; k += 4) {
      v2f a, b;
      a.x = A[arow][k + koff];
      a.y = A[arow][k + koff + 1];
      b.x = Bp[(k + koff) * DP + bcol];
      b.y = Bp[(k + koff + 1) * DP + bcol];
      acc = __builtin_amdgcn_wmma_f32_16x16x4_f32(false, a, false, b,
                                                  (short)0, acc, false, false);
    }
    const int rb = mt * 16 + ((lane >> 4) << 3);
    const int cc = n0 + (lane & 15);
#pragma unroll
    for (int v = 0; v < 8; ++v) Dst[rb + v][cc] = acc[v];
  }
}

// -----------------------------------------------------------------------------
// Kernel 1: persistent single-workgroup sequential scan over 128 timesteps.
// -----------------------------------------------------------------------------
__global__ __launch_bounds__(1024) void rgcn_scan_kernel(
    const int* __restrict__ x_indices,   // [S, NN]
    const int* __restrict__ edge_index,  // [S, 2, EE]
    const int* __restrict__ edge_type,   // [S, EE]
    const float* __restrict__ X,         // [NUM_NODES, DD]
    const float* __restrict__ Wpad,      // [5, DD, DP] padded weights
    const float* __restrict__ Wg,        // [NN*DD, DD]
    const float* __restrict__ Ug,        // [DD, DD]
    float* __restrict__ memG,            // ws: [NN, DD]
    float* __restrict__ x1s)             // ws: [S, DD]
{
  __shared__ __attribute__((aligned(16))) float gx[NN][DP];
  __shared__ float xwr[NN][DP];    // per-relation transform
  __shared__ float comp[NN][DP];   // composite / proposed accumulator
  __shared__ float uacc[DP];       // gate pre-activation / gate
  __shared__ float dis[RR][NN];    // rsqrt(deg)
  __shared__ int   degc[RR][NN];
  __shared__ int   se[EE], de[EE], te[EE];
  __shared__ int   idxs[NN];

  const int tid = threadIdx.x;
  const int lane = tid & 31;
  const int wid = tid >> 5;
  const int nthr = blockDim.x;

  // mem0 = zeros (deterministic for every graph replay); zero gx pad columns once
  for (int p = tid; p < NN * DD; p += nthr) memG[p] = 0.0f;
  if (tid < NN * 4) gx[tid >> 2][DD + (tid & 3)] = 0.0f;
  __threadfence_block();
  __syncthreads();

  for (int step = 0; step < S_TOT; ++step) {
    // ---- load indices / edges, clear degrees ----
    if (tid < NN) idxs[tid] = x_indices[step * NN + tid];
    if (tid < EE) {
      se[tid] = edge_index[step * 2 * EE + tid];
      de[tid] = edge_index[step * 2 * EE + EE + tid];
      te[tid] = edge_type[step * EE + tid];
    }
    if (tid < RR * NN) ((int*)degc)[tid] = 0;
    __syncthreads();

    // ---- gather gx = X[idx] : async global->LDS (128b), ASYNCcnt tracked ----
    {
      const int row_q = DD / 4;  // 75 x vec4 per row, 16B aligned both sides
      for (int p = tid; p < NN * row_q; p += nthr) {
        int n = p / row_q, d = (p % row_q) * 4;
        const float* g = X + (long)idxs[n] * DD + d;
#ifdef USE_ASYNC_GATHER
        __builtin_amdgcn_global_load_async_to_lds_b128(
            (AS1 i32x4*)g, (AS3 i32x4*)&gx[n][d], 0, 0);
#else
        *(v4f*)&gx[n][d] = *(const v4f*)g;
#endif
      }
    }
    if (tid < EE) atomicAdd(&degc[te[tid]][de[tid]], 1);
#ifdef USE_ASYNC_GATHER
    __builtin_amdgcn_s_wait_asynccnt(0);
#endif
    __syncthreads();
    if (tid < RR * NN) {
      int r = tid / NN, n = tid % NN;
      dis[r][n] = rsqrtf((float)degc[r][n] + 1.0f);
    }
    __syncthreads();

    // ---- comp = gx @ W_0 (self term of "proposed") ----
    gemm_tiles_f32(gx, Wpad + 4 * DD * DP, comp, lane, wid);
    __syncthreads();

    // ---- per-relation RGCN conv ----
    for (int r = 0; r < RR; ++r) {
      gemm_tiles_f32(gx, Wpad + (long)r * DD * DP, xwr, lane, wid);
      __syncthreads();
      // edge messages: comp[dst] += xw[src] * dis[src]*dis[dst]
      for (int p = tid; p < EE * DD; p += nthr) {
        int e = p / DD, d = p % DD;
        if (te[e] == r) {
          float w = dis[r][se[e]] * dis[r][de[e]];
          atomicAdd(&comp[de[e]][d], xwr[se[e]][d] * w);
        }
      }
      __syncthreads();
      // self-loop messages: comp[n] += xw[n] * dis[n]^2
      for (int p = tid; p < NN * DD; p += nthr) {
        int n = p / DD, d = p % DD;
        float s = dis[r][n];
        comp[n][d] += xwr[n][d] * (s * s);
      }
      __syncthreads();
    }

    // ---- GRU update gate: u = sigmoid(gx.flatten() @ Wg + mem[0] @ Ug) ----
    if (tid < DP) uacc[tid] = 0.0f;
    __syncthreads();
    if (tid < DD * 3) {                   // 3 K-slices of 3200 per column
      const int d = tid % DD;             // consecutive tids -> coalesced Wg
      const int s = tid / DD;
      float partial = 0.0f;
      const int i0 = s * 3200;
      int n = i0 / DD, k = i0 % DD;
      long wgi = (long)i0 * DD + d;
      for (int c = 0; c < 3200; ++c) {
        partial += gx[n][k] * Wg[wgi];
        wgi += DD;
        if (++k == DD) { k = 0; ++n; }
      }
      if (s == 0) {
        for (int kk = 0; kk < DD; ++kk)
          partial += memG[kk] * Ug[kk * DD + d];   // mem row 0
      }
      atomicAdd(&uacc[d], partial);
    }
    __syncthreads();
    if (tid < DD) uacc[tid] = 1.0f / (1.0f + expf(-uacc[tid]));
    __syncthreads();

    // ---- new = u*proposed + (1-u)*mem ; x1 = leaky_relu(new[0]) ----
    for (int p = tid; p < NN * DD; p += nthr) {
      int n = p / DD, d = p % DD;
      float u = uacc[d];
      float m = memG[p];
      float nv = u * comp[n][d] + (1.0f - u) * m;
      memG[p] = nv;
      if (n == 0) x1s[step * DD + d] = (nv > 0.0f) ? nv : 0.01f * nv;
    }
    __threadfence_block();
    __syncthreads();
  }
}

// -----------------------------------------------------------------------------
// Kernel 2: logits[S,V] = x1s[S,DD] @ lin_w[V,DD]^T + lin_b, fp32 WMMA.
// V is tile-exact (50000 = 3125*16): column validity is wave-uniform, so the
// hot loop carries no guards and EXEC is all-ones for every live wave.
// -----------------------------------------------------------------------------
__global__ __launch_bounds__(256) void logits_gemm_kernel(
    const float* __restrict__ x1s, const float* __restrict__ lin_w,
    const float* __restrict__ lin_b, float* __restrict__ logits) {
  __shared__ float a_sh[16][DP];
  const int tid = threadIdx.x;
  const int lane = tid & 31;
  const int wid = tid >> 5;                    // 0..7 -> one 16-col tile/wave
  const int s0 = blockIdx.y * 16;
  const int col_base = blockIdx.x * 128;

  for (int p = tid; p < 16 * DD; p += blockDim.x) {
    int rr = p / DD, k = p % DD;
    a_sh[rr][k] = x1s[(s0 + rr) * DD + k];
  }
  __syncthreads();

  if (col_base + wid * 16 >= VV) return;       // wave-uniform tail exit

  const int c = col_base + wid * 16 + (lane & 15);
  const long boff = (long)c * DD;
  const int koff = (lane >> 4) << 1;
  const int arow = lane & 15;
  v8f acc = {};
  for (int k = 0; k < DD; k += 4) {
    v2f a, b;
    a.x = a_sh[arow][k + koff];
    a.y = a_sh[arow][k + koff + 1];
    b.x = lin_w[boff + k + koff];
    b.y = lin_w[boff + k + koff + 1];
    acc = __builtin_amdgcn_wmma_f32_16x16x4_f32(false, a, false, b,
                                                (short)0, acc, false, false);
  }
  const float bias = lin_b[c];
  const int rb = (lane >> 4) << 3;
#pragma unroll
  for (int v = 0; v < 8; ++v)
    logits[(long)(s0 + rb + v) * VV + c] = acc[v] + bias;
}

// -----------------------------------------------------------------------------
// Kernel 3: row-wise log_softmax into d_out, plus zero preds_senses tail.
// -----------------------------------------------------------------------------
__global__ __launch_bounds__(256) void logsoftmax_kernel(
    const float* __restrict__ logits, float* __restrict__ out) {
  __shared__ float red[256];
  const int s = blockIdx.x;
  const int tid = threadIdx.x;
  const float* row = logits + (long)s * VV;

  float m = -3.4e38f;
  for (int v = tid; v < VV; v += 256) m = fmaxf(m, row[v]);
  red[tid] = m;
  __syncthreads();
  for (int o = 128; o > 0; o >>= 1) {
    if (tid < o) red[tid] = fmaxf(red[tid], red[tid + o]);
    __syncthreads();
  }
  m = red[0];
  __syncthreads();

  float sum = 0.0f;
  for (int v = tid; v < VV; v += 256) sum += expf(row[v] - m);
  red[tid] = sum;
  __syncthreads();
  for (int o = 128; o > 0; o >>= 1) {
    if (tid < o) red[tid] += red[tid + o];
    __syncthreads();
  }
  const float lse = m + logf(red[0]);

  for (int v = tid; v < VV; v += 256)
    out[(long)s * VV + v] = row[v] - lse;
  if (tid == 0) out[(long)S_TOT * VV + s] = 0.0f;  // preds_senses == 0
}

// -----------------------------------------------------------------------------
extern "C" void kernel_launch(void* const* d_in, const int* in_sizes, int n_in,
                              void* d_out, int out_size, void* d_ws, size_t ws_size,
                              hipStream_t stream) {
  (void)in_sizes; (void)n_in; (void)out_size; (void)ws_size;
  const int*   x_indices  = (const int*)d_in[0];
  const int*   edge_index = (const int*)d_in[1];
  const int*   edge_type  = (const int*)d_in[2];
  const float* X          = (const float*)d_in[3];
  const float* conv_W     = (const float*)d_in[4];
  const float* W_0        = (const float*)d_in[5];
  const float* Wg         = (const float*)d_in[6];
  const float* Ug         = (const float*)d_in[7];
  const float* lin_w      = (const float*)d_in[8];
  const float* lin_b      = (const float*)d_in[9];
  float* out = (float*)d_out;

  // ws layout (floats): mem[9600] | x1s[38400] | Wpad[5*300*304] | logits[128*50000]
  float* ws    = (float*)d_ws;
  float* memG  = ws;
  float* x1s   = ws + NN * DD;                       // 9600
  float* Wpad  = ws + NN * DD + S_TOT * DD;          // 48000
  float* logit = Wpad + 5 * DD * DP;                 // 504000

  pad_weights_kernel<<<(5 * DD * DP + 255) / 256, 256, 0, stream>>>(conv_W, W_0, Wpad);

  rgcn_scan_kernel<<<1, 1024, 0, stream>>>(x_indices, edge_index, edge_type, X,
                                           Wpad, Wg, Ug, memG, x1s);

  dim3 g2((VV + 127) / 128, S_TOT / 16);             // (391, 8)
  logits_gemm_kernel<<<g2, 256, 0, stream>>>(x1s, lin_w, lin_b, logit);

  logsoftmax_kernel<<<S_TOT, 256, 0, stream>>>(logit, out);
}